// GatedGCNLayer_39187281608763
// MI455X (gfx1250) — compile-verified
//
#include <hip/hip_runtime.h>
#include <hip/hip_bf16.h>

typedef __attribute__((ext_vector_type(2))) float v2f;
typedef __attribute__((ext_vector_type(8))) float v8f;

#define DD 96
#define NT6 6

// ---------------------------------------------------------------- zero scratch
__global__ void ggcn_zero_kernel(float* __restrict__ p, long n) {
    long i = (long)blockIdx.x * blockDim.x + threadIdx.x;
    long stride = (long)gridDim.x * blockDim.x;
    for (; i < n; i += stride) p[i] = 0.0f;
}

// ------------------------------------------------- node GEMMs: Ah,Bh,Dh,Eh
// One block = 4 waves; wave w computes out_w = h[16 rows] @ W_w^T + b_w via
// V_WMMA_F32_16X16X4_F32.  A-layout: a[v] = X[m][k0+v+2*lanehi]; B mirrored.
__global__ __launch_bounds__(128) void ggcn_node_gemm_kernel(
    const float* __restrict__ h,
    const float* __restrict__ wA, const float* __restrict__ bA,
    const float* __restrict__ wB, const float* __restrict__ bB,
    const float* __restrict__ wD, const float* __restrict__ bD,
    const float* __restrict__ wE, const float* __restrict__ bE,
    float* __restrict__ outA, float* __restrict__ outB,
    float* __restrict__ outD, float* __restrict__ outE, int nrows) {
    const int wave   = threadIdx.x >> 5;
    const int lane   = threadIdx.x & 31;
    const int lanelo = lane & 15;
    const int lanehi = lane >> 4;
    const int mtile  = blockIdx.x;

    const float* W; const float* bias; float* out;
    switch (wave) {
        case 0:  W = wA; bias = bA; out = outA; break;
        case 1:  W = wB; bias = bB; out = outB; break;
        case 2:  W = wD; bias = bD; out = outD; break;
        default: W = wE; bias = bE; out = outE; break;
    }

    int arow = mtile * 16 + lanelo;
    if (arow >= nrows) arow = nrows - 1;   // clamp loads; stores guarded below

    v8f acc[NT6];
#pragma unroll
    for (int nt = 0; nt < NT6; ++nt) acc[nt] = (v8f)0.0f;

    for (int k0 = 0; k0 < DD; k0 += 4) {
        const int ka = k0 + 2 * lanehi;
        v2f a = *(const v2f*)(h + (long)arow * DD + ka);
#pragma unroll
        for (int nt = 0; nt < NT6; ++nt) {
            const int col = nt * 16 + lanelo;
            v2f b = *(const v2f*)(W + (long)col * DD + ka);
            acc[nt] = __builtin_amdgcn_wmma_f32_16x16x4_f32(
                false, a, false, b, (short)0, acc[nt], false, false);
        }
    }

#pragma unroll
    for (int nt = 0; nt < NT6; ++nt) {
        const int col = nt * 16 + lanelo;
        const float bv = bias[col];
#pragma unroll
        for (int r = 0; r < 8; ++r) {
            const int row = mtile * 16 + r + 8 * lanehi;
            if (row < nrows) out[(long)row * DD + col] = acc[nt][r] + bv;
        }
    }
}

// ------------------------------------------------- edge kernel
// Per wave: 16-edge tile.  Ce = e_tile @ C_w^T (WMMA f32), then
// e_new = Ce + C_b + Dh[src] + Eh[dst];  sigma = sigmoid(e_new);
// store e_new (pre-BN) to d_out e-section; atomic segment sums into L2.
__global__ __launch_bounds__(256) void ggcn_edge_kernel(
    const float* __restrict__ e, const int* __restrict__ src,
    const int* __restrict__ dst, const float* __restrict__ wC,
    const float* __restrict__ bC, const float* __restrict__ Bh,
    const float* __restrict__ Dh, const float* __restrict__ Eh,
    float* __restrict__ ssh, float* __restrict__ ss,
    float* __restrict__ enew_out, int nedges) {
    const int wave   = threadIdx.x >> 5;
    const int lane   = threadIdx.x & 31;
    const int lanelo = lane & 15;
    const int lanehi = lane >> 4;
    const int etile  = blockIdx.x * 8 + wave;
    const long base  = (long)etile * 16;
    if (base >= nedges) return;          // wave-uniform exit (EXEC stays full)

    int arow = (int)base + lanelo;
    if (arow >= nedges) arow = nedges - 1;

    v8f acc[NT6];
#pragma unroll
    for (int nt = 0; nt < NT6; ++nt) acc[nt] = (v8f)0.0f;

    for (int k0 = 0; k0 < DD; k0 += 4) {
        const int ka = k0 + 2 * lanehi;
        v2f a = *(const v2f*)(e + (long)arow * DD + ka);
#pragma unroll
        for (int nt = 0; nt < NT6; ++nt) {
            const int col = nt * 16 + lanelo;
            v2f b = *(const v2f*)(wC + (long)col * DD + ka);
            acc[nt] = __builtin_amdgcn_wmma_f32_16x16x4_f32(
                false, a, false, b, (short)0, acc[nt], false, false);
        }
    }

#pragma unroll
    for (int r = 0; r < 8; ++r) {
        const long row = base + r + 8 * lanehi;
        if (row >= nedges) continue;
        const long s = (long)src[row] * DD;
        const long d = (long)dst[row] * DD;
#pragma unroll
        for (int nt = 0; nt < NT6; ++nt) {
            const int col = nt * 16 + lanelo;
            const float v = acc[nt][r] + bC[col] + Dh[s + col] + Eh[d + col];
            const float sg = 1.0f / (1.0f + __expf(-v));
            enew_out[row * DD + col] = v;
            unsafeAtomicAdd(&ssh[d + col], Bh[s + col] * sg);
            unsafeAtomicAdd(&ss[d + col], sg);
        }
    }
}

// ------------------------------------------------- h_new + per-column stats
// blockDim.x == 96 (one thread per feature column); rows strided by grid.
__global__ void ggcn_hnew_stats_kernel(const float* __restrict__ Ah,
                                       const float* __restrict__ ssh,
                                       const float* __restrict__ ss,
                                       float* __restrict__ hnew,
                                       float* __restrict__ sum,
                                       float* __restrict__ sumsq, int nrows) {
    const int col = threadIdx.x;
    float ls = 0.0f, lq = 0.0f;
    for (int row = blockIdx.x; row < nrows; row += gridDim.x) {
        const long i = (long)row * DD + col;
        const float v = Ah[i] + ssh[i] / (ss[i] + 1e-6f);
        hnew[i] = v;
        ls += v;
        lq += v * v;
    }
    unsafeAtomicAdd(&sum[col], ls);
    unsafeAtomicAdd(&sumsq[col], lq);
}

// ------------------------------------------------- column stats of e_new
__global__ void ggcn_colstats_kernel(const float* __restrict__ x,
                                     float* __restrict__ sum,
                                     float* __restrict__ sumsq, long nrows) {
    const int col = threadIdx.x;
    float ls = 0.0f, lq = 0.0f;
    for (long row = blockIdx.x; row < nrows; row += gridDim.x) {
        const float v = x[row * DD + col];
        ls += v;
        lq += v * v;
    }
    unsafeAtomicAdd(&sum[col], ls);
    unsafeAtomicAdd(&sumsq[col], lq);
}

// ------------------------------------------------- BN + ReLU + residual
__global__ void ggcn_bn_relu_res_kernel(const float* __restrict__ xin,
                                        float* __restrict__ xbuf,
                                        const float* __restrict__ sum,
                                        const float* __restrict__ sumsq,
                                        const float* __restrict__ gamma,
                                        const float* __restrict__ beta,
                                        long nelem, float invn) {
    long i = (long)blockIdx.x * blockDim.x + threadIdx.x;
    const long stride = (long)gridDim.x * blockDim.x;
    for (; i < nelem; i += stride) {
        const int col = (int)(i % DD);
        const float m = sum[col] * invn;
        const float var = sumsq[col] * invn - m * m;
        const float x = xbuf[i];
        float y = gamma[col] * (x - m) * rsqrtf(var + 1e-5f) + beta[col];
        y = fmaxf(y, 0.0f);
        xbuf[i] = xin[i] + y;
    }
}

extern "C" void kernel_launch(void* const* d_in, const int* in_sizes, int n_in,
                              void* d_out, int out_size, void* d_ws,
                              size_t ws_size, hipStream_t stream) {
    const float* h    = (const float*)d_in[0];
    const float* e    = (const float*)d_in[1];
    const int*   src  = (const int*)d_in[2];
    const int*   dst  = (const int*)d_in[3];
    const float* A_w  = (const float*)d_in[4];
    const float* A_b  = (const float*)d_in[5];
    const float* B_w  = (const float*)d_in[6];
    const float* B_b  = (const float*)d_in[7];
    const float* C_w  = (const float*)d_in[8];
    const float* C_b  = (const float*)d_in[9];
    const float* Dl_w = (const float*)d_in[10];
    const float* Dl_b = (const float*)d_in[11];
    const float* El_w = (const float*)d_in[12];
    const float* El_b = (const float*)d_in[13];
    const float* gamma_h = (const float*)d_in[14];
    const float* beta_h  = (const float*)d_in[15];
    const float* gamma_e = (const float*)d_in[16];
    const float* beta_e  = (const float*)d_in[17];

    const int N = in_sizes[0] / DD;
    const int E = in_sizes[1] / DD;
    const long ND = (long)N * DD;
    const long ED = (long)E * DD;

    // workspace layout (floats)
    float* ws   = (float*)d_ws;
    float* Ah   = ws;
    float* Bh   = ws + 1 * ND;
    float* Dh   = ws + 2 * ND;
    float* Eh   = ws + 3 * ND;
    float* ssh  = ws + 4 * ND;  // segment_sum(Bh[src]*sigma)
    float* ss   = ws + 5 * ND;  // segment_sum(sigma)
    float* st   = ws + 6 * ND;  // [sum_h | sumsq_h | sum_e | sumsq_e], 4*96
    float* sum_h   = st;
    float* sumsq_h = st + DD;
    float* sum_e   = st + 2 * DD;
    float* sumsq_e = st + 3 * DD;

    float* out_h = (float*)d_out;         // h_out: N*96
    float* out_e = out_h + ND;            // e_out: E*96

    // 1) zero accumulators (segment sums + stats)
    ggcn_zero_kernel<<<2048, 256, 0, stream>>>(ssh, 2 * ND + 4 * DD);

    // 2) node GEMMs (WMMA f32 16x16x4)
    const int mtiles = (N + 15) / 16;
    ggcn_node_gemm_kernel<<<mtiles, 128, 0, stream>>>(
        h, A_w, A_b, B_w, B_b, Dl_w, Dl_b, El_w, El_b, Ah, Bh, Dh, Eh, N);

    // 3) edge GEMM + gather + sigmoid + atomic segment sums; e_new -> out_e
    const int etiles = (E + 15) / 16;
    ggcn_edge_kernel<<<(etiles + 7) / 8, 256, 0, stream>>>(
        e, src, dst, C_w, C_b, Bh, Dh, Eh, ssh, ss, out_e, E);

    // 4) h_new (pre-BN) -> out_h, and h column stats
    ggcn_hnew_stats_kernel<<<512, DD, 0, stream>>>(Ah, ssh, ss, out_h, sum_h,
                                                   sumsq_h, N);

    // 5) e_new column stats
    ggcn_colstats_kernel<<<2048, DD, 0, stream>>>(out_e, sum_e, sumsq_e,
                                                  (long)E);

    // 6) BN + ReLU + residual, in place on d_out
    ggcn_bn_relu_res_kernel<<<4096, 256, 0, stream>>>(
        h, out_h, sum_h, sumsq_h, gamma_h, beta_h, ND, 1.0f / (float)N);
    ggcn_bn_relu_res_kernel<<<8192, 256, 0, stream>>>(
        e, out_e, sum_e, sumsq_e, gamma_e, beta_e, ED, 1.0f / (float)E);
}